// _rpn_89979564851943
// MI455X (gfx1250) — compile-verified
//
#include <hip/hip_runtime.h>
#include <math.h>

// ---------------------------------------------------------------- constants
#define FHW   80
#define NPIX  6400        // 80*80
#define CIN   256
#define COUT  256
#define NB    4
#define KTOT  (CIN*9)     // 2304
#define NANCH 57600       // 6400*9
#define PRE_NMS  1000
#define POST_NMS 300
#define SORT_N   65536    // next pow2 >= 57600
#define IMG_WH   1280.0f

typedef __attribute__((ext_vector_type(2))) float v2f;
typedef __attribute__((ext_vector_type(8))) float v8f;

// ---------------------------------------------------------------- helpers
__device__ __forceinline__ float bn_affine(float x, float g, float b, float m, float v) {
    float inv = g / sqrtf(v + 1e-5f);
    return x * inv + (b - m * inv);
}
__device__ __forceinline__ float mishf(float x) {
    // softplus = max(x,0) + log1p(exp(-|x|))  (stable, matches jax.nn.softplus)
    float sp = fmaxf(x, 0.f) + log1pf(expf(-fabsf(x)));
    return x * tanhf(sp);
}

// ---------------------------------------------------------------- kernel 0:
// one-time weight transpose [co][ci][tap] -> [co][tap][ci]  (tap-major K)
__global__ void transpose_w(const float* __restrict__ w, float* __restrict__ wtr)
{
    int t = blockIdx.x * 256 + threadIdx.x;      // 0 .. 256*2304-1
    if (t >= COUT * KTOT) return;
    int co  = t / KTOT;
    int rem = t - co * KTOT;
    int tap = rem >> 8;          // /256
    int ci  = rem & 255;
    wtr[t] = w[co * KTOT + ci * 9 + tap];
}

// ---------------------------------------------------------------- kernel A:
// 3x3 conv + bias + mish + BN1 via V_WMMA_F32_16X16X4_F32 implicit GEMM.
// Register-blocked: one wave -> 32(co) x 32(pix) = 4 accumulators; per K-step
// 2 A-fragments + 2 B-fragments feed 4 WMMAs (2x fragment reuse each).
// K reordered tap-major (K=tap*256+ci); padding via clamped addr + 0/1 mask.
__global__ __launch_bounds__(128)
void conv_mish_bn_wmma(const float* __restrict__ x, const float* __restrict__ wtr,
                       const float* __restrict__ bias,
                       const float* __restrict__ g1, const float* __restrict__ b1,
                       const float* __restrict__ m1, const float* __restrict__ v1,
                       float* __restrict__ basef)
{
    const int lane = threadIdx.x & 31;
    const int wave = threadIdx.x >> 5;
    const int ml   = lane & 15;          // row/col within 16
    const int kh   = lane >> 4;          // K-half selector (0/1)
    const int coBase = blockIdx.y * 32;                 // 8 co-pairs
    const int pt     = (blockIdx.x * 4 + wave) * 2;     // 200 pixel-tile pairs
    const int n      = blockIdx.z;
    const int pix0 = pt * 16 + ml;       // each 16-tile is row-aligned (80%16==0)
    const int pix1 = pix0 + 16;
    const int py0 = pix0 / FHW, px0 = pix0 - py0 * FHW;
    const int py1 = pix1 / FHW, px1 = pix1 - py1 * FHW;

    const float* Wrow0 = wtr + (size_t)(coBase + ml) * KTOT;        // [co][tap][ci]
    const float* Wrow1 = Wrow0 + (size_t)16 * KTOT;
    const float* Xn    = x + (size_t)n * CIN * NPIX;

    v8f acc00 = {0.f,0.f,0.f,0.f,0.f,0.f,0.f,0.f};   // co[0:16)  x pix0
    v8f acc01 = acc00;                               // co[0:16)  x pix1
    v8f acc10 = acc00;                               // co[16:32) x pix0
    v8f acc11 = acc00;                               // co[16:32) x pix1

    #pragma unroll
    for (int tap = 0; tap < 9; ++tap) {
        const int ky = tap / 3 - 1;
        const int kx = tap - (tap / 3) * 3 - 1;
        const int yy0 = py0 + ky, xx0 = px0 + kx;
        const int yy1 = py1 + ky, xx1 = px1 + kx;
        const float msk0 = (((unsigned)yy0 < (unsigned)FHW) && ((unsigned)xx0 < (unsigned)FHW)) ? 1.f : 0.f;
        const float msk1 = (((unsigned)yy1 < (unsigned)FHW) && ((unsigned)xx1 < (unsigned)FHW)) ? 1.f : 0.f;
        const float* Xp0 = Xn + min(max(yy0,0),FHW-1) * FHW + min(max(xx0,0),FHW-1);
        const float* Xp1 = Xn + min(max(yy1,0),FHW-1) * FHW + min(max(xx1,0),FHW-1);
        const float* Wt0 = Wrow0 + tap * CIN;
        const float* Wt1 = Wrow1 + tap * CIN;
        #pragma unroll 2
        for (int ci = 0; ci < CIN; ci += 4) {
            const int c = ci + 2 * kh;                   // this lane's K pair
            v2f a0 = *(const v2f*)(Wt0 + c);             // b64 loads
            v2f a1 = *(const v2f*)(Wt1 + c);
            v2f b0, b1;
            b0.x = Xp0[(size_t)c       * NPIX] * msk0;   // branchless padding
            b0.y = Xp0[(size_t)(c + 1) * NPIX] * msk0;
            b1.x = Xp1[(size_t)c       * NPIX] * msk1;
            b1.y = Xp1[(size_t)(c + 1) * NPIX] * msk1;
            acc00 = __builtin_amdgcn_wmma_f32_16x16x4_f32(false, a0, false, b0, (short)0, acc00, false, false);
            acc01 = __builtin_amdgcn_wmma_f32_16x16x4_f32(false, a0, false, b1, (short)0, acc01, false, false);
            acc10 = __builtin_amdgcn_wmma_f32_16x16x4_f32(false, a1, false, b0, (short)0, acc10, false, false);
            acc11 = __builtin_amdgcn_wmma_f32_16x16x4_f32(false, a1, false, b1, (short)0, acc11, false, false);
        }
    }
    // epilogue: C layout -> lane holds D[M=r+8*kh][N=ml] per 16x16 sub-tile
    #pragma unroll
    for (int r = 0; r < 8; ++r) {
        int coA = coBase + r + 8 * kh;
        int coB = coA + 16;
        float vA = bn_affine(mishf(acc00[r] + bias[coA]), g1[coA], b1[coA], m1[coA], v1[coA]);
        float vB = bn_affine(mishf(acc01[r] + bias[coA]), g1[coA], b1[coA], m1[coA], v1[coA]);
        float vC = bn_affine(mishf(acc10[r] + bias[coB]), g1[coB], b1[coB], m1[coB], v1[coB]);
        float vD = bn_affine(mishf(acc11[r] + bias[coB]), g1[coB], b1[coB], m1[coB], v1[coB]);
        basef[((size_t)n * COUT + coA) * NPIX + pix0] = vA;
        basef[((size_t)n * COUT + coA) * NPIX + pix1] = vB;
        basef[((size_t)n * COUT + coB) * NPIX + pix0] = vC;
        basef[((size_t)n * COUT + coB) * NPIX + pix1] = vD;
    }
}

// ---------------------------------------------------------------- kernel B:
// 1x1 heads: per (n, pixel, anchor) compute 2 cls + 4 reg dots over 256ch,
// BN affines, softmax pair -> fg; write fg and rpn_reg to d_out.
__global__ __launch_bounds__(256)
void heads_kernel(const float* __restrict__ basef,
                  const float* __restrict__ clsw, const float* __restrict__ regw,
                  const float* __restrict__ g2, const float* __restrict__ b2,
                  const float* __restrict__ m2, const float* __restrict__ v2,
                  const float* __restrict__ g3, const float* __restrict__ b3,
                  const float* __restrict__ m3, const float* __restrict__ v3,
                  float* __restrict__ out_fg, float* __restrict__ out_reg)
{
    const int p = blockIdx.x * 256 + threadIdx.x;  // pixel
    const int a = blockIdx.y;                      // anchor 0..8
    const int n = blockIdx.z;
    const float* B  = basef + (size_t)n * COUT * NPIX;
    const float* c0 = clsw + (size_t)(2 * a) * CIN;       // uniform -> scalar loads
    const float* c1 = c0 + CIN;
    const float* r0 = regw + (size_t)(4 * a) * CIN;
    float s0=0.f, s1=0.f, d0=0.f, d1=0.f, d2=0.f, d3=0.f;
    for (int ci = 0; ci < CIN; ++ci) {
        float xb = B[ci * NPIX + p];               // coalesced across p
        s0 += xb * c0[ci];          s1 += xb * c1[ci];
        d0 += xb * r0[ci];          d1 += xb * r0[CIN + ci];
        d2 += xb * r0[2*CIN + ci];  d3 += xb * r0[3*CIN + ci];
    }
    int c2a = 2 * a, c4a = 4 * a;
    s0 = bn_affine(s0, g2[c2a],   b2[c2a],   m2[c2a],   v2[c2a]);
    s1 = bn_affine(s1, g2[c2a+1], b2[c2a+1], m2[c2a+1], v2[c2a+1]);
    d0 = bn_affine(d0, g3[c4a],   b3[c4a],   m3[c4a],   v3[c4a]);
    d1 = bn_affine(d1, g3[c4a+1], b3[c4a+1], m3[c4a+1], v3[c4a+1]);
    d2 = bn_affine(d2, g3[c4a+2], b3[c4a+2], m3[c4a+2], v3[c4a+2]);
    d3 = bn_affine(d3, g3[c4a+3], b3[c4a+3], m3[c4a+3], v3[c4a+3]);
    float fg = 1.f / (1.f + expf(s0 - s1));        // softmax([s0,s1])[1]
    size_t ai = (size_t)n * NANCH + (size_t)p * 9 + a;
    out_fg[ai] = fg;
    out_reg[ai*4+0] = d0; out_reg[ai*4+1] = d1;
    out_reg[ai*4+2] = d2; out_reg[ai*4+3] = d3;
}

// ---------------------------------------------------------------- kernel C:
// anchor gen + decode + clip + min-size filter
__global__ void decode_kernel(const float* __restrict__ fg, const float* __restrict__ reg,
                              float* __restrict__ boxesAll, float* __restrict__ scoresAll)
{
    int t = blockIdx.x * 256 + threadIdx.x;
    if (t >= NB * NANCH) return;
    int n = t / NANCH; int i = t - n * NANCH;
    int p = i / 9;     int a = i - p * 9;
    int py = p / FHW;  int px = p - py * FHW;
    int ir = a / 3;    int is = a - ir * 3;
    const float ratios[3] = {0.5f, 1.0f, 2.0f};
    const float scales[3] = {8.f, 16.f, 32.f};
    float r = ratios[ir], s = scales[is];
    float ws = rintf(sqrtf(1024.f / r));   // rintf = round half-even (np.round)
    float hs = rintf(ws * r);
    float Wb = ws * s, Hb = hs * s;
    float sx = px * 16.f, sy = py * 16.f;
    float ax1 = 15.5f - 0.5f*(Wb-1.f) + sx, ay1 = 15.5f - 0.5f*(Hb-1.f) + sy;
    float ax2 = 15.5f + 0.5f*(Wb-1.f) + sx, ay2 = 15.5f + 0.5f*(Hb-1.f) + sy;
    float aw = ax2-ax1+1.f, ah = ay2-ay1+1.f;
    float acx = ax1 + 0.5f*aw, acy = ay1 + 0.5f*ah;
    const float* d = reg + (size_t)t * 4;
    float pcx = d[0]*aw + acx,  pcy = d[1]*ah + acy;
    float pw  = expf(d[2])*aw,  ph  = expf(d[3])*ah;
    float x1 = fminf(fmaxf(pcx - 0.5f*pw, 0.f), IMG_WH - 1.f);
    float y1 = fminf(fmaxf(pcy - 0.5f*ph, 0.f), IMG_WH - 1.f);
    float x2 = fminf(fmaxf(pcx + 0.5f*pw, 0.f), IMG_WH - 1.f);
    float y2 = fminf(fmaxf(pcy + 0.5f*ph, 0.f), IMG_WH - 1.f);
    boxesAll[(size_t)t*4+0] = x1; boxesAll[(size_t)t*4+1] = y1;
    boxesAll[(size_t)t*4+2] = x2; boxesAll[(size_t)t*4+3] = y2;
    float bw = x2-x1+1.f, bh = y2-y1+1.f;
    scoresAll[t] = (bw >= 16.f && bh >= 16.f) ? fg[t] : -1e9f;
}

// ---------------------------------------------------------------- top-k via
// stable bitonic sort (score desc, index asc) on padded 65536 per image.
__global__ void sort_init(const float* __restrict__ scoresAll,
                          float* __restrict__ keys, int* __restrict__ idxs)
{
    int t = blockIdx.x * 256 + threadIdx.x;   // 0 .. 4*65536-1
    int n = t >> 16; int i = t & (SORT_N - 1);
    float k = -3.0e38f; int id = 0x7fffffff;
    if (i < NANCH) { k = scoresAll[n * NANCH + i]; id = i; }
    keys[t] = k; idxs[t] = id;
}
__device__ __forceinline__ bool pair_gt(float sa, int ia, float sb, int ib) {
    return (sa > sb) || (sa == sb && ia < ib);
}
__global__ void bitonic_pass(float* __restrict__ keys, int* __restrict__ idxs,
                             int kstep, int jstep)
{
    int t = blockIdx.x * 256 + threadIdx.x;
    int n = t >> 16; int i = t & (SORT_N - 1);
    int l = i ^ jstep;
    if (l <= i) return;
    size_t bo = (size_t)n << 16;
    float si = keys[bo+i], sl = keys[bo+l];
    int   ii = idxs[bo+i], il = idxs[bo+l];
    bool dirDesc = ((i & kstep) == 0);
    if (pair_gt(sl, il, si, ii) == dirDesc) {
        keys[bo+i] = sl; keys[bo+l] = si;
        idxs[bo+i] = il; idxs[bo+l] = ii;
    }
}

// ---------------------------------------------------------------- kernel E:
// sequential NMS on top-1000 in LDS, then emit kept boxes (score order) + zeros
__global__ __launch_bounds__(256)
void nms_kernel(const float* __restrict__ boxesAll, const int* __restrict__ idxs,
                float* __restrict__ rois)
{
    __shared__ float bx[PRE_NMS][4];
    __shared__ float area[PRE_NMS];
    __shared__ int   supp[PRE_NMS];
    const int n = blockIdx.x, tid = threadIdx.x;
    const int* id = idxs + ((size_t)n << 16);
    for (int i = tid; i < PRE_NMS; i += 256) {
        int bi = id[i];
        const float* b = boxesAll + ((size_t)n * NANCH + bi) * 4;
        float x1=b[0], y1=b[1], x2=b[2], y2=b[3];
        bx[i][0]=x1; bx[i][1]=y1; bx[i][2]=x2; bx[i][3]=y2;
        area[i] = (x2-x1+1.f)*(y2-y1+1.f);
        supp[i] = 0;
    }
    __syncthreads();
    for (int i = 0; i < PRE_NMS; ++i) {
        if (!supp[i]) {
            float x1=bx[i][0], y1=bx[i][1], x2=bx[i][2], y2=bx[i][3], ar=area[i];
            for (int j = i + 1 + tid; j < PRE_NMS; j += 256) {
                float xx1 = fmaxf(x1, bx[j][0]);
                float yy1 = fmaxf(y1, bx[j][1]);
                float xx2 = fminf(x2, bx[j][2]);
                float yy2 = fminf(y2, bx[j][3]);
                float inter = fmaxf(xx2-xx1+1.f, 0.f) * fmaxf(yy2-yy1+1.f, 0.f);
                float iou = inter / (ar + area[j] - inter);
                if (iou > 0.7f) supp[j] = 1;
            }
        }
        __syncthreads();
    }
    float* R = rois + (size_t)n * POST_NMS * 4;
    if (tid == 0) {
        int cnt = 0;
        for (int i = 0; i < PRE_NMS && cnt < POST_NMS; ++i) {
            if (!supp[i]) {
                R[cnt*4+0]=bx[i][0]; R[cnt*4+1]=bx[i][1];
                R[cnt*4+2]=bx[i][2]; R[cnt*4+3]=bx[i][3];
                ++cnt;
            }
        }
        for (; cnt < POST_NMS; ++cnt) {
            R[cnt*4+0]=0.f; R[cnt*4+1]=0.f; R[cnt*4+2]=0.f; R[cnt*4+3]=0.f;
        }
    }
}

// ---------------------------------------------------------------- launcher
extern "C" void kernel_launch(void* const* d_in, const int* in_sizes, int n_in,
                              void* d_out, int out_size, void* d_ws, size_t ws_size,
                              hipStream_t stream)
{
    (void)in_sizes; (void)n_in; (void)out_size; (void)ws_size;
    const float* x     = (const float*)d_in[0];
    const float* convw = (const float*)d_in[1];
    const float* convb = (const float*)d_in[2];
    const float* g1 = (const float*)d_in[3];  const float* b1 = (const float*)d_in[4];
    const float* m1 = (const float*)d_in[5];  const float* v1 = (const float*)d_in[6];
    const float* clsw = (const float*)d_in[7];
    const float* g2 = (const float*)d_in[8];  const float* b2 = (const float*)d_in[9];
    const float* m2 = (const float*)d_in[10]; const float* v2 = (const float*)d_in[11];
    const float* regw = (const float*)d_in[12];
    const float* g3 = (const float*)d_in[13]; const float* b3 = (const float*)d_in[14];
    const float* m3 = (const float*)d_in[15]; const float* v3 = (const float*)d_in[16];

    float* out      = (float*)d_out;
    float* out_fg   = out;                                  // 4*57600
    float* out_reg  = out + (size_t)NB * NANCH;             // 4*57600*4
    float* out_rois = out + (size_t)NB * NANCH * 5;         // 4*300*4

    float* ws        = (float*)d_ws;
    float* basef     = ws;                                  // 4*256*6400   = 6,553,600 f
    float* boxesAll  = ws + 6553600;                        // 4*57600*4    =   921,600 f
    float* scoresAll = boxesAll + (size_t)NB * NANCH * 4;   // 4*57600      =   230,400 f
    float* keys      = scoresAll + (size_t)NB * NANCH;      // 4*65536      =   262,144 f
    int*   idxsp     = (int*)(keys + (size_t)NB * SORT_N);  // 4*65536 ints
    float* wtr       = (float*)(idxsp + (size_t)NB * SORT_N); // 256*2304 = 589,824 f

    transpose_w<<<dim3((COUT * KTOT + 255) / 256), 256, 0, stream>>>(convw, wtr);

    conv_mish_bn_wmma<<<dim3(50, 8, NB), 128, 0, stream>>>(
        x, wtr, convb, g1, b1, m1, v1, basef);

    heads_kernel<<<dim3(NPIX / 256, 9, NB), 256, 0, stream>>>(
        basef, clsw, regw, g2, b2, m2, v2, g3, b3, m3, v3, out_fg, out_reg);

    decode_kernel<<<dim3((NB * NANCH + 255) / 256), 256, 0, stream>>>(
        out_fg, out_reg, boxesAll, scoresAll);

    sort_init<<<dim3(NB * SORT_N / 256), 256, 0, stream>>>(scoresAll, keys, idxsp);
    for (int k = 2; k <= SORT_N; k <<= 1)
        for (int j = k >> 1; j > 0; j >>= 1)
            bitonic_pass<<<dim3(NB * SORT_N / 256), 256, 0, stream>>>(keys, idxsp, k, j);

    nms_kernel<<<dim3(NB), 256, 0, stream>>>(boxesAll, idxsp, out_rois);
}